// StandardGCNEncoder_67156108640278
// MI455X (gfx1250) — compile-verified
//
#include <hip/hip_runtime.h>

typedef float v2f __attribute__((ext_vector_type(2)));
typedef float v8f __attribute__((ext_vector_type(8)));

#define N_NODES 50000
#define FDIM 64

#define WMMA_F32(a, b, c) \
    __builtin_amdgcn_wmma_f32_16x16x4_f32(false, (a), false, (b), (short)0, (c), false, false)

// ---------------- degree kernels ----------------
__global__ void gcn_deg_init(float* __restrict__ deg, int n) {
    int i = blockIdx.x * blockDim.x + threadIdx.x;
    if (i < n) deg[i] = 1.0f;  // self-loop contributes 1
}

__global__ void gcn_deg_count(const int* __restrict__ dst, float* __restrict__ deg, int E) {
    int e = blockIdx.x * blockDim.x + threadIdx.x;
    if (e < E) unsafeAtomicAdd(&deg[dst[e]], 1.0f);
}

__global__ void gcn_deg_rsqrt(float* __restrict__ deg, int n) {
    int i = blockIdx.x * blockDim.x + threadIdx.x;
    if (i < n) deg[i] = rsqrtf(deg[i]);  // deg >= 1 always
}

// ---------------- fused fp32 WMMA GEMM strip ----------------
// H[M,64] = act(A[M,K]) * W[K,64]   (act = ReLU if RELU_A)
// OutInit[m,f] = H[m,f] * dinv[m]^2 + bias[f]   (self-loop + bias epilogue)
//
// W is staged in LDS once per block, shared by all 8 waves. LDS layout puts
// each column's k-quad contiguous: lds[(k>>2)*256 + 4n + (k&3)], so a wave's
// B-fragment ds_load_b64 at 4*lo + 2*hi covers all 64 banks exactly once
// (conflict-free), with strip offsets 0/64/128/192 floats as immediates.
// One wave32 computes a 16x64 strip with 4 accumulators; the A stream stays
// on global memory and is software-pipelined one k-step ahead.
template <int K, bool RELU_A>
__global__ void gcn_gemm_fused(const float* __restrict__ A,
                               const float* __restrict__ W,  // [K][64] row-major
                               const float* __restrict__ dinv,
                               const float* __restrict__ bias,
                               float* __restrict__ H,
                               float* __restrict__ OutInit,
                               int M) {
    __shared__ float bsh[K * 64];

    const int tid = threadIdx.x;
    // cooperative, coalesced load of W into LDS (pair-quad interleaved layout)
    for (int i = tid; i < K * 64; i += 256) {
        int k = i >> 6;
        int n = i & 63;
        bsh[(k >> 2) * 256 + 4 * n + (k & 3)] = W[i];
    }
    __syncthreads();

    const int wave = tid >> 5;
    const int lane = tid & 31;
    const int tileM = blockIdx.x * 8 + wave;
    if (tileM * 16 >= M) return;  // wave-uniform exit (EXEC all-1s inside)

    const int hi = lane >> 4;   // 0: K pair {0,1}; 1: K pair {2,3}
    const int lo = lane & 15;

    const float* arow  = A + (long long)(tileM * 16 + lo) * K;  // A row m = lo
    const float* bbase = bsh + 4 * lo + 2 * hi;                 // + strip*64 + (kk>>2)*256

    v8f c0 = {}, c1 = {}, c2 = {}, c3 = {};

    v2f a = *(const v2f*)(arow + 2 * hi);  // k-step 0 fragment

    for (int kk = 0; kk < K; kk += 4) {
        v2f an;
        if (kk + 4 < K) an = *(const v2f*)(arow + kk + 4 + 2 * hi);  // prefetch next

        const float* bp = bbase + (kk >> 2) * 256;
        v2f b0 = *(const v2f*)(bp + 0);
        v2f b1 = *(const v2f*)(bp + 64);
        v2f b2 = *(const v2f*)(bp + 128);
        v2f b3 = *(const v2f*)(bp + 192);

        if (RELU_A) { a.x = fmaxf(a.x, 0.0f); a.y = fmaxf(a.y, 0.0f); }
        c0 = WMMA_F32(a, b0, c0);
        c1 = WMMA_F32(a, b1, c1);
        c2 = WMMA_F32(a, b2, c2);
        c3 = WMMA_F32(a, b3, c3);

        a = an;
    }

    // epilogue: C/D layout -> row m = tileM*16 + 8*hi + v, col = strip*16 + lo
    const int rowBase = tileM * 16 + 8 * hi;
    const float bb0 = bias[lo];
    const float bb1 = bias[16 + lo];
    const float bb2 = bias[32 + lo];
    const float bb3 = bias[48 + lo];
#pragma unroll
    for (int v = 0; v < 8; ++v) {
        const long long roff = (long long)(rowBase + v) * FDIM + lo;
        float dv = dinv[rowBase + v];
        float dd = dv * dv;
        float* hr  = H + roff;
        float* orr = OutInit + roff;
        hr[0]  = c0[v];  orr[0]  = c0[v] * dd + bb0;
        hr[16] = c1[v];  orr[16] = c1[v] * dd + bb1;
        hr[32] = c2[v];  orr[32] = c2[v] * dd + bb2;
        hr[48] = c3[v];  orr[48] = c3[v] * dd + bb3;
    }
}

// ---------------- edge aggregation: out[dst] += h[src] * dinv[src]*dinv[dst] ----------------
// 16 lanes per edge, float4 (b128) gather + 4 hardware fp32 atomics per lane.
__global__ void gcn_edge_agg(const int* __restrict__ src,
                             const int* __restrict__ dst,
                             const float* __restrict__ dinv,
                             const float4* __restrict__ h4,  // h viewed as [N][16] float4
                             float* __restrict__ out, int E) {
    long long tid = (long long)blockIdx.x * blockDim.x + threadIdx.x;
    long long e = tid >> 4;
    int q = (int)(tid & 15);
    if (e >= E) return;
    int s = src[e];
    int d = dst[e];
    float w = dinv[s] * dinv[d];
    float4 v = h4[(long long)s * 16 + q];
    float* o = out + (long long)d * FDIM + q * 4;
    unsafeAtomicAdd(o + 0, v.x * w);
    unsafeAtomicAdd(o + 1, v.y * w);
    unsafeAtomicAdd(o + 2, v.z * w);
    unsafeAtomicAdd(o + 3, v.w * w);
}

// ---------------- launcher ----------------
extern "C" void kernel_launch(void* const* d_in, const int* in_sizes, int n_in,
                              void* d_out, int out_size, void* d_ws, size_t ws_size,
                              hipStream_t stream) {
    const float* x  = (const float*)d_in[0];
    const int*   ei = (const int*)d_in[1];     // [2, E]
    const float* W1 = (const float*)d_in[2];   // [256, 64]
    const float* b1 = (const float*)d_in[3];
    const float* W2 = (const float*)d_in[4];   // [64, 64]
    const float* b2 = (const float*)d_in[5];
    float* out = (float*)d_out;

    const int E  = in_sizes[1] / 2;
    const int N  = N_NODES;
    const int NF = N * FDIM;                   // 3.2M

    const int* src = ei;
    const int* dst = ei + E;

    // Workspace (floats): dinv[50048] | h1[NF] | agg1[NF] | h2[NF]
    float* ws   = (float*)d_ws;
    float* dinv = ws;
    float* h1   = ws + 50048;                  // 16B-aligned
    float* agg1 = h1 + NF;
    float* h2   = agg1 + NF;

    const int T = 256;
    // degrees -> dinv
    gcn_deg_init  <<<(N + T - 1) / T, T, 0, stream>>>(dinv, N);
    gcn_deg_count <<<(E + T - 1) / T, T, 0, stream>>>(dst, dinv, E);
    gcn_deg_rsqrt <<<(N + T - 1) / T, T, 0, stream>>>(dinv, N);

    const int mTiles = N / 16;                 // 3125
    const int gblk   = (mTiles + 7) / 8;       // 8 waves/block, one 16x64 strip each

    const long long eThreads = (long long)E * 16;  // 16 lanes per edge
    const int eBlocks = (int)((eThreads + T - 1) / T);

    // layer 1: h1 = x @ W1 ; agg1 init = h1*dinv^2 + b1 ; agg1 += edges
    gcn_gemm_fused<256, false> <<<gblk, T, 0, stream>>>(x, W1, dinv, b1, h1, agg1, N);
    gcn_edge_agg               <<<eBlocks, T, 0, stream>>>(src, dst, dinv, (const float4*)h1, agg1, E);

    // layer 2: h2 = relu(agg1) @ W2 ; out init = h2*dinv^2 + b2 ; out += edges
    gcn_gemm_fused<64, true>   <<<gblk, T, 0, stream>>>(agg1, W2, dinv, b2, h2, out, N);
    gcn_edge_agg               <<<eBlocks, T, 0, stream>>>(src, dst, dinv, (const float4*)h2, out, E);
}